// SSMDecoder_31937376813583
// MI455X (gfx1250) — compile-verified
//
#include <hip/hip_runtime.h>
#include <math.h>

// ---------------------------------------------------------------------------
// SSM decoder for MI455X (gfx1250, wave32).
// GEMMs: V_WMMA_F32_16X16X4_F32 (exact fp32, matches the f32 reference),
// 128x64 block tile, 2x2 wave tiles, LDS staging via GLOBAL_LOAD_ASYNC_TO_LDS
// (ASYNCcnt) when the toolchain exposes the gfx1250 builtin. Scan: VALU
// kernel, 16-wide SSM state in registers per (b,d) lane.
// ---------------------------------------------------------------------------

typedef float v2f __attribute__((ext_vector_type(2)));
typedef float v8f __attribute__((ext_vector_type(8)));
typedef int   v4i __attribute__((ext_vector_type(4)));

#if defined(__has_builtin)
#if __has_builtin(__builtin_amdgcn_global_load_async_to_lds_b128) && \
    __has_builtin(__builtin_amdgcn_s_wait_asynccnt)
#define USE_ASYNC_LDS 1
#endif
#endif

#define BB 8
#define TT 1024
#define SS 128
#define MM 512
#define DD 512
#define NN 16
#define LLAY 3
#define NAT 8
#define MVAR 20
#define BT (BB * TT)

#define TM 128
#define TN 64
#define TK 32
#define LDSW 36   // row stride (floats): 16B-aligned rows + conflict-free frags

// ---------------------------------------------------------------------------
// C[m,n] = sum_k A[m,k] * W[n,k] + bias[n]     (x @ W^T + b), batched via z.
// ---------------------------------------------------------------------------
__global__ __launch_bounds__(256)
void gemm_wmma_f32(const float* __restrict__ A, long long lda, long long sAb,
                   const float* __restrict__ W, long long ldw, long long sWb,
                   float* __restrict__ C, long long ldc, long long sCb,
                   const float* __restrict__ bias,
                   int Mtot, int Ntot, int K)
{
    __shared__ __align__(16) float As[TM][LDSW];
    __shared__ __align__(16) float Bs[TN][LDSW];

    const int tid  = threadIdx.x;
    const int w    = tid >> 5;
    const int wm   = w & 3;        // 4 wave bands of 32 rows
    const int wn   = w >> 2;       // 2 wave bands of 32 cols
    const int lane = tid & 31;
    const int half = lane >> 4;    // K-half per ISA 7.12.2 f32 A/B layout
    const int lr   = lane & 15;

    const int m0 = blockIdx.y * TM;
    const int n0 = blockIdx.x * TN;

    A += (long long)blockIdx.z * sAb;
    W += (long long)blockIdx.z * sWb;
    C += (long long)blockIdx.z * sCb;

    const bool interior = (m0 + TM <= Mtot) && (n0 + TN <= Ntot); // block-uniform

    v8f acc00 = {}, acc01 = {}, acc10 = {}, acc11 = {};

    for (int k0 = 0; k0 < K; k0 += TK) {
        if (interior) {
#if defined(USE_ASYNC_LDS)
            // gfx1250 async copy: GLOBAL_LOAD_ASYNC_TO_LDS_B128 (ASYNCcnt)
            #pragma unroll
            for (int j = 0; j < 4; ++j) {
                int idx = tid + j * 256;          // 1024 x 16B for A slab
                int m = idx >> 3, kq = idx & 7;
                __builtin_amdgcn_global_load_async_to_lds_b128(
                    (v4i*)(A + (long long)(m0 + m) * lda + k0 + kq * 4),
                    (v4i*)&As[m][kq * 4], 0, 0);
            }
            #pragma unroll
            for (int j = 0; j < 2; ++j) {
                int idx = tid + j * 256;          // 512 x 16B for W slab
                int n = idx >> 3, kq = idx & 7;
                __builtin_amdgcn_global_load_async_to_lds_b128(
                    (v4i*)(W + (long long)(n0 + n) * ldw + k0 + kq * 4),
                    (v4i*)&Bs[n][kq * 4], 0, 0);
            }
            __builtin_amdgcn_s_wait_asynccnt(0);
#else
            #pragma unroll
            for (int j = 0; j < 4; ++j) {
                int idx = tid + j * 256;
                int m = idx >> 3, kq = idx & 7;
                const float4 v = *(const float4*)(A + (long long)(m0 + m) * lda + k0 + kq * 4);
                *(float4*)&As[m][kq * 4] = v;
            }
            #pragma unroll
            for (int j = 0; j < 2; ++j) {
                int idx = tid + j * 256;
                int n = idx >> 3, kq = idx & 7;
                const float4 v = *(const float4*)(W + (long long)(n0 + n) * ldw + k0 + kq * 4);
                *(float4*)&Bs[n][kq * 4] = v;
            }
#endif
        } else {
            // edge blocks: scalar guarded staging
            #pragma unroll
            for (int j = 0; j < 16; ++j) {
                int idx = tid + j * 256;          // TM*TK = 4096
                int m = idx >> 5, kk = idx & 31;
                int gm = m0 + m;
                As[m][kk] = (gm < Mtot) ? A[(long long)gm * lda + k0 + kk] : 0.f;
            }
            #pragma unroll
            for (int j = 0; j < 8; ++j) {
                int idx = tid + j * 256;          // TN*TK = 2048
                int n = idx >> 5, kk = idx & 31;
                int gn = n0 + n;
                Bs[n][kk] = (gn < Ntot) ? W[(long long)gn * ldw + k0 + kk] : 0.f;
            }
        }
        __syncthreads();

        // prefetch next A K-slab while the matrix pipe works
        if (interior && (k0 + TK < K)) {
            int pm = m0 + (tid >> 1);
            __builtin_prefetch(A + (long long)pm * lda + k0 + TK + (tid & 1) * 16, 0, 1);
        }

        const int ar = wm * 32 + lr;
        const int br = wn * 32 + lr;
        #pragma unroll
        for (int kk = 0; kk < TK; kk += 4) {
            int kb = kk + 2 * half;
            v2f a0, a1, b0, b1;
            a0.x = As[ar][kb];       a0.y = As[ar][kb + 1];
            a1.x = As[ar + 16][kb];  a1.y = As[ar + 16][kb + 1];
            b0.x = Bs[br][kb];       b0.y = Bs[br][kb + 1];
            b1.x = Bs[br + 16][kb];  b1.y = Bs[br + 16][kb + 1];
            acc00 = __builtin_amdgcn_wmma_f32_16x16x4_f32(false, a0, false, b0, (short)0, acc00, false, false);
            acc01 = __builtin_amdgcn_wmma_f32_16x16x4_f32(false, a0, false, b1, (short)0, acc01, false, false);
            acc10 = __builtin_amdgcn_wmma_f32_16x16x4_f32(false, a1, false, b0, (short)0, acc10, false, false);
            acc11 = __builtin_amdgcn_wmma_f32_16x16x4_f32(false, a1, false, b1, (short)0, acc11, false, false);
        }
        __syncthreads();
    }

    // C layout: VGPR r -> M = r + 8*laneHalf, N = lane%16 (per 16x16 tile)
    const int colA = n0 + wn * 32 + lr;
    const int colB = colA + 16;
    if (interior) {
        #pragma unroll
        for (int r = 0; r < 8; ++r) {
            int rowA = m0 + wm * 32 + half * 8 + r;
            int rowB = rowA + 16;
            float b0v = bias ? bias[colA] : 0.f;
            float b1v = bias ? bias[colB] : 0.f;
            C[(long long)rowA * ldc + colA] = acc00[r] + b0v;
            C[(long long)rowA * ldc + colB] = acc01[r] + b1v;
            C[(long long)rowB * ldc + colA] = acc10[r] + b0v;
            C[(long long)rowB * ldc + colB] = acc11[r] + b1v;
        }
    } else {
        #pragma unroll
        for (int r = 0; r < 8; ++r) {
            int rowA = m0 + wm * 32 + half * 8 + r;
            int rowB = rowA + 16;
            if (rowA < Mtot && colA < Ntot)
                C[(long long)rowA * ldc + colA] = acc00[r] + (bias ? bias[colA] : 0.f);
            if (rowA < Mtot && colB < Ntot)
                C[(long long)rowA * ldc + colB] = acc01[r] + (bias ? bias[colB] : 0.f);
            if (rowB < Mtot && colA < Ntot)
                C[(long long)rowB * ldc + colA] = acc10[r] + (bias ? bias[colA] : 0.f);
            if (rowB < Mtot && colB < Ntot)
                C[(long long)rowB * ldc + colB] = acc11[r] + (bias ? bias[colB] : 0.f);
        }
    }
}

// ---------------------------------------------------------------------------
// Decoder-input construction
// ---------------------------------------------------------------------------
__global__ __launch_bounds__(256)
void gather_sym_kernel(const float* __restrict__ sym,
                       const int* __restrict__ tg,
                       float* __restrict__ U)
{
    long long i = (long long)blockIdx.x * 256 + threadIdx.x;   // BT*DD
    int d = (int)(i & (DD - 1));
    long long row = i >> 9;
    int b = (int)(row >> 10);
    int t = (int)(row & (TT - 1));
    int g = t ? tg[b * TT + t - 1] : 0;
    int gi = g < 0 ? 0 : (g > SS - 1 ? SS - 1 : g);
    U[row * DD + d] = sym[((long long)b * SS + gi) * DD + d];
}

__global__ __launch_bounds__(256)
void build_input_kernel(const float* __restrict__ action_embed,
                        const float* __restrict__ P,        // ptr_emb (bias folded in)
                        const float* __restrict__ var_embed,
                        const int* __restrict__ ta,
                        const int* __restrict__ tg,
                        float* __restrict__ out)            // BT x 2D concat
{
    long long i = (long long)blockIdx.x * 256 + threadIdx.x;
    int d = (int)(i & (DD - 1));
    long long row = i >> 9;
    int b = (int)(row >> 10);
    int t = (int)(row & (TT - 1));
    int a = t ? ta[b * TT + t - 1] : 0;   // END_CLAUSE at t=0
    int g = t ? tg[b * TT + t - 1] : 0;
    float arg;
    if (a == 1 || a == 2) {               // PRED | ARG_FUNC -> pointer embed
        arg = P[row * DD + d];
    } else if (a == 3) {                  // ARG_VAR
        int gi = g < 0 ? 0 : (g > MVAR - 1 ? MVAR - 1 : g);
        arg = var_embed[gi * DD + d];
    } else {
        arg = 0.f;
    }
    out[row * (2 * DD) + d]      = action_embed[a * DD + d];
    out[row * (2 * DD) + DD + d] = arg;
}

__global__ __launch_bounds__(256)
void mem_mean_kernel(const float* __restrict__ mem, float* __restrict__ out)
{
    int i = blockIdx.x * 256 + threadIdx.x;   // B*D = 4096
    int b = i >> 9, d = i & (DD - 1);
    float s = 0.f;
    for (int m = 0; m < MM; ++m)
        s += mem[((long long)b * MM + m) * DD + d];
    out[i] = s * (1.f / MM);
}

// ---------------------------------------------------------------------------
// SSM pieces
// ---------------------------------------------------------------------------
__global__ __launch_bounds__(256)
void softplus_kernel(float* __restrict__ x)
{
    long long i = (long long)blockIdx.x * 256 + threadIdx.x;
    float v = x[i];
    x[i] = (v > 20.f) ? v : log1pf(__expf(v));
}

// One lane per (b,d); SSM state h[0..15] lives in VGPRs; sequential over T.
__global__ __launch_bounds__(256)
void scan_kernel(const float* __restrict__ xz,   // BT x 2D (x_ssm = cols 0..D)
                 const float* __restrict__ dt,   // BT x D
                 const float* __restrict__ Bm,   // BT x N
                 const float* __restrict__ Cm,   // BT x N
                 const float* __restrict__ Alog, // D x N (this layer)
                 float* __restrict__ y)          // BT x D
{
    int gid = blockIdx.x * 256 + threadIdx.x;    // 0..4095
    int b = gid >> 9;
    int d = gid & (DD - 1);

    float Areg[NN], h[NN];
    #pragma unroll
    for (int n = 0; n < NN; ++n) { Areg[n] = -__expf(Alog[d * NN + n]); h[n] = 0.f; }

    const float* xrow = xz + (long long)b * TT * (2 * DD) + d;
    const float* drow = dt + (long long)b * TT * DD + d;
    const float* Brow = Bm + (long long)b * TT * NN;
    const float* Crow = Cm + (long long)b * TT * NN;
    float*       yrow = y  + (long long)b * TT * DD + d;

    for (int t = 0; t < TT; ++t) {
        float x   = xrow[(long long)t * (2 * DD)];
        float dtv = drow[(long long)t * DD];
        float dx  = dtv * x;
        float acc = 0.f;
        #pragma unroll
        for (int n = 0; n < NN; ++n) {
            float dA = __expf(Areg[n] * dtv);          // v_exp_f32 (TRANS, co-exec)
            h[n] = dA * h[n] + dx * Brow[t * NN + n];
            acc += h[n] * Crow[t * NN + n];
        }
        yrow[(long long)t * DD] = acc;
    }
}

__global__ __launch_bounds__(256)
void ycombine_kernel(float* __restrict__ y, const float* __restrict__ xz,
                     const float* __restrict__ Dp)
{
    long long i = (long long)blockIdx.x * 256 + threadIdx.x;   // BT*D
    int d = (int)(i & (DD - 1));
    long long row = i >> 9;
    float x = xz[row * (2 * DD) + d];
    float z = xz[row * (2 * DD) + DD + d];
    float sz = z / (1.f + __expf(-z));                          // silu
    y[i] = (y[i] + Dp[d] * x) * sz;
}

// ---------------------------------------------------------------------------
// Fused LayerNorms (one block per row, D=512, 2 elems/thread)
// ---------------------------------------------------------------------------
__global__ __launch_bounds__(256)
void ln_residual_kernel(float* __restrict__ h, const float* __restrict__ y2,
                        const float* __restrict__ g, const float* __restrict__ bta)
{
    int row = blockIdx.x, tid = threadIdx.x;
    const float* hr = h  + (long long)row * DD;
    const float* yr = y2 + (long long)row * DD;
    float r0 = hr[tid] + yr[tid];
    float r1 = hr[tid + 256] + yr[tid + 256];
    __shared__ float s1[256], s2[256];
    s1[tid] = r0 + r1; s2[tid] = r0 * r0 + r1 * r1;
    __syncthreads();
    for (int st = 128; st > 0; st >>= 1) {
        if (tid < st) { s1[tid] += s1[tid + st]; s2[tid] += s2[tid + st]; }
        __syncthreads();
    }
    float mean = s1[0] * (1.f / DD);
    float var  = s2[0] * (1.f / DD) - mean * mean;
    float inv  = rsqrtf(var + 1e-5f);
    float* ho = h + (long long)row * DD;
    ho[tid]       = (r0 - mean) * inv * g[tid] + bta[tid];
    ho[tid + 256] = (r1 - mean) * inv * g[tid + 256] + bta[tid + 256];
}

__global__ __launch_bounds__(256)
void ln_gate_kernel(float* __restrict__ h, const float* __restrict__ gp,
                    const float* __restrict__ ctx,
                    const float* __restrict__ g, const float* __restrict__ bta)
{
    int row = blockIdx.x, tid = threadIdx.x;
    int b = row >> 10;
    const float* hr = h   + (long long)row * DD;
    const float* gr = gp  + (long long)row * DD;
    const float* cr = ctx + (long long)b * DD;
    float g0 = 1.f / (1.f + __expf(-gr[tid]));
    float g1 = 1.f / (1.f + __expf(-gr[tid + 256]));
    float r0 = hr[tid]       + g0 * cr[tid];
    float r1 = hr[tid + 256] + g1 * cr[tid + 256];
    __shared__ float s1[256], s2[256];
    s1[tid] = r0 + r1; s2[tid] = r0 * r0 + r1 * r1;
    __syncthreads();
    for (int st = 128; st > 0; st >>= 1) {
        if (tid < st) { s1[tid] += s1[tid + st]; s2[tid] += s2[tid + st]; }
        __syncthreads();
    }
    float mean = s1[0] * (1.f / DD);
    float var  = s2[0] * (1.f / DD) - mean * mean;
    float inv  = rsqrtf(var + 1e-5f);
    float* ho = h + (long long)row * DD;
    ho[tid]       = (r0 - mean) * inv * g[tid] + bta[tid];
    ho[tid + 256] = (r1 - mean) * inv * g[tid + 256] + bta[tid + 256];
}

__global__ __launch_bounds__(256)
void concat_hctx_kernel(const float* __restrict__ h, const float* __restrict__ ctx,
                        float* __restrict__ out)
{
    long long i = (long long)blockIdx.x * 256 + threadIdx.x;   // BT*D
    int d = (int)(i & (DD - 1));
    long long row = i >> 9;
    int b = (int)(row >> 10);
    out[row * (2 * DD) + d]      = h[i];
    out[row * (2 * DD) + DD + d] = ctx[(long long)b * DD + d];
}

// ---------------------------------------------------------------------------
// Attention-score epilogue: scale by 1/sqrt(D), mask to -inf
// ---------------------------------------------------------------------------
__global__ __launch_bounds__(256)
void mask_scores_kernel(float* __restrict__ sc, const unsigned char* __restrict__ mask)
{
    long long i = (long long)blockIdx.x * 256 + threadIdx.x;   // B*T*S
    int s = (int)(i & (SS - 1));
    int b = (int)(i >> 17);                                    // T*S = 2^17
    float v = sc[i] * 0.04419417382415922f;                    // 1/sqrt(512)
    sc[i] = mask[b * SS + s] ? v : -__builtin_inff();
}

// ---------------------------------------------------------------------------
// Host
// ---------------------------------------------------------------------------
extern "C" void kernel_launch(void* const* d_in, const int* in_sizes, int n_in,
                              void* d_out, int out_size, void* d_ws, size_t ws_size,
                              hipStream_t stream)
{
    const float* symbol_embeds = (const float*)d_in[0];
    const float* memory        = (const float*)d_in[1];
    const float* action_embed  = (const float*)d_in[2];
    const float* W_arg         = (const float*)d_in[3];
    const float* b_arg         = (const float*)d_in[4];
    const float* var_embed     = (const float*)d_in[5];
    const float* W_comb        = (const float*)d_in[6];
    const float* b_comb        = (const float*)d_in[7];
    const float* Win           = (const float*)d_in[8];
    const float* b_in          = (const float*)d_in[9];
    const float* Wdt           = (const float*)d_in[10];
    const float* b_dt          = (const float*)d_in[11];
    const float* Alog          = (const float*)d_in[12];
    const float* WB            = (const float*)d_in[13];
    const float* b_B           = (const float*)d_in[14];
    const float* WC            = (const float*)d_in[15];
    const float* b_C           = (const float*)d_in[16];
    const float* Dp            = (const float*)d_in[17];
    const float* Wout          = (const float*)d_in[18];
    const float* b_out         = (const float*)d_in[19];
    const float* ssm_g         = (const float*)d_in[20];
    const float* ssm_b         = (const float*)d_in[21];
    const float* Wm            = (const float*)d_in[22];
    const float* b_m           = (const float*)d_in[23];
    const float* Wg            = (const float*)d_in[24];
    const float* b_g           = (const float*)d_in[25];
    const float* cn_g          = (const float*)d_in[26];
    const float* cn_b          = (const float*)d_in[27];
    const float* Wa            = (const float*)d_in[28];
    const float* b_a           = (const float*)d_in[29];
    const float* Wq            = (const float*)d_in[30];
    const float* b_q           = (const float*)d_in[31];
    const float* Wk            = (const float*)d_in[32];
    const float* b_k           = (const float*)d_in[33];
    const float* Wv            = (const float*)d_in[34];
    const float* b_v           = (const float*)d_in[35];
    const int*   target_actions   = (const int*)d_in[36];
    const int*   target_arguments = (const int*)d_in[37];
    const unsigned char* symbol_mask = (const unsigned char*)d_in[38];

    // workspace layout (floats); buffers reused across phases
    float* ws = (float*)d_ws;
    size_t off = 0;
    auto take = [&](size_t n) { float* p = ws + off; off += n; return p; };
    float* buf_h   = take((size_t)BT * DD);        // running hidden state
    float* buf2    = take((size_t)BT * 2 * DD);    // concat / xz
    float* buf_dt  = take((size_t)BT * DD);        // ptr_emb -> dt -> y2 -> gate -> k
    float* buf_y   = take((size_t)BT * DD);        // gathered syms -> scan y -> q
    float* buf_Bm  = take((size_t)BT * NN);
    float* buf_Cm  = take((size_t)BT * NN);
    float* buf_ms  = take((size_t)BB * DD);
    float* buf_ctx = take((size_t)BB * DD);
    (void)ws_size; (void)n_in; (void)in_sizes; (void)out_size;

    float* outF      = (float*)d_out;
    float* out_act   = outF;                                   // (B,T,NA)
    float* out_sc    = outF + (size_t)BT * NAT;                // (B,T,S)
    float* out_var   = out_sc + (size_t)BB * TT * SS;          // (B,T,MV)

    auto gemm = [&](const float* A, long long lda, long long sAb,
                    const float* W, long long ldw, long long sWb,
                    float* Cp, long long ldc, long long sCb,
                    const float* bias, int M, int N, int K, int batch) {
        dim3 grid((N + TN - 1) / TN, (M + TM - 1) / TM, batch);
        gemm_wmma_f32<<<grid, dim3(256), 0, stream>>>(
            A, lda, sAb, W, ldw, sWb, Cp, ldc, sCb, bias, M, N, K);
    };

    const int EW = (BT * DD) / 256;   // elementwise grid over BT*D

    // ---- decoder-input construction --------------------------------------
    gather_sym_kernel<<<EW, 256, 0, stream>>>(symbol_embeds, target_arguments, buf_y);
    gemm(buf_y, DD, 0, W_arg, DD, 0, buf_dt, DD, 0, b_arg, BT, DD, DD, 1);   // ptr_emb
    build_input_kernel<<<EW, 256, 0, stream>>>(action_embed, buf_dt, var_embed,
                                               target_actions, target_arguments, buf2);
    gemm(buf2, 2 * DD, 0, W_comb, 2 * DD, 0, buf_h, DD, 0, b_comb, BT, DD, 2 * DD, 1);
    mem_mean_kernel<<<(BB * DD) / 256, 256, 0, stream>>>(memory, buf_ms);

    // ---- 3x (SSM block + cross-gate) -------------------------------------
    for (int l = 0; l < LLAY; ++l) {
        const float* Win_l  = Win  + (size_t)l * 2 * DD * DD;
        const float* bin_l  = b_in + (size_t)l * 2 * DD;
        const float* Wdt_l  = Wdt  + (size_t)l * DD * DD;
        const float* bdt_l  = b_dt + (size_t)l * DD;
        const float* Alog_l = Alog + (size_t)l * DD * NN;
        const float* WB_l   = WB   + (size_t)l * NN * DD;
        const float* bB_l   = b_B  + (size_t)l * NN;
        const float* WC_l   = WC   + (size_t)l * NN * DD;
        const float* bC_l   = b_C  + (size_t)l * NN;
        const float* Dp_l   = Dp   + (size_t)l * DD;
        const float* Wout_l = Wout + (size_t)l * DD * DD;
        const float* bout_l = b_out+ (size_t)l * DD;
        const float* sg_l   = ssm_g+ (size_t)l * DD;
        const float* sb_l   = ssm_b+ (size_t)l * DD;
        const float* Wm_l   = Wm   + (size_t)l * DD * DD;
        const float* bm_l   = b_m  + (size_t)l * DD;
        const float* Wg_l   = Wg   + (size_t)l * DD * 2 * DD;
        const float* bg_l   = b_g  + (size_t)l * DD;
        const float* cg_l   = cn_g + (size_t)l * DD;
        const float* cb_l   = cn_b + (size_t)l * DD;

        // xz = h @ Win^T + b_in   (BT x 2D)
        gemm(buf_h, DD, 0, Win_l, DD, 0, buf2, 2 * DD, 0, bin_l, BT, 2 * DD, DD, 1);
        // dt = softplus(x_ssm @ Wdt^T + b_dt)
        gemm(buf2, 2 * DD, 0, Wdt_l, DD, 0, buf_dt, DD, 0, bdt_l, BT, DD, DD, 1);
        softplus_kernel<<<EW, 256, 0, stream>>>(buf_dt);
        // B,C projections (N=16)
        gemm(buf2, 2 * DD, 0, WB_l, DD, 0, buf_Bm, NN, 0, bB_l, BT, NN, DD, 1);
        gemm(buf2, 2 * DD, 0, WC_l, DD, 0, buf_Cm, NN, 0, bC_l, BT, NN, DD, 1);
        // selective scan (sequential over T, 4096 parallel lanes)
        scan_kernel<<<(BB * DD) / 256, 256, 0, stream>>>(buf2, buf_dt, buf_Bm,
                                                         buf_Cm, Alog_l, buf_y);
        // y = (y + Dp*x) * silu(z)
        ycombine_kernel<<<EW, 256, 0, stream>>>(buf_y, buf2, Dp_l);
        // y2 = y @ Wout^T + b_out ; h = LN(y2 + h)
        gemm(buf_y, DD, 0, Wout_l, DD, 0, buf_dt, DD, 0, bout_l, BT, DD, DD, 1);
        ln_residual_kernel<<<BT, 256, 0, stream>>>(buf_h, buf_dt, sg_l, sb_l);
        // ctx = mem_summary @ Wm^T + b_m  (M=8)
        gemm(buf_ms, DD, 0, Wm_l, DD, 0, buf_ctx, DD, 0, bm_l, BB, DD, DD, 1);
        // gate = sigmoid([h|ctx] @ Wg^T + b_g) ; h = LN(h + gate*ctx)
        concat_hctx_kernel<<<EW, 256, 0, stream>>>(buf_h, buf_ctx, buf2);
        gemm(buf2, 2 * DD, 0, Wg_l, 2 * DD, 0, buf_dt, DD, 0, bg_l, BT, DD, 2 * DD, 1);
        ln_gate_kernel<<<BT, 256, 0, stream>>>(buf_h, buf_dt, buf_ctx, cg_l, cb_l);
    }

    // ---- heads ------------------------------------------------------------
    gemm(buf_h, DD, 0, Wa, DD, 0, out_act, NAT, 0, b_a, BT, NAT, DD, 1);
    gemm(buf_h, DD, 0, Wq, DD, 0, buf_y, DD, 0, b_q, BT, DD, DD, 1);           // q
    gemm(symbol_embeds, DD, 0, Wk, DD, 0, buf_dt, DD, 0, b_k, BB * SS, DD, DD, 1); // k
    // scores[b] = q[b] @ k[b]^T  (batched over grid.z)
    gemm(buf_y, DD, (long long)TT * DD,
         buf_dt, DD, (long long)SS * DD,
         out_sc, SS, (long long)TT * SS,
         nullptr, TT, SS, DD, BB);
    mask_scores_kernel<<<(BB * TT * SS) / 256, 256, 0, stream>>>(out_sc, symbol_mask);
    gemm(buf_h, DD, 0, Wv, DD, 0, out_var, MVAR, 0, b_v, BT, MVAR, DD, 1);
}